// two_layer_GAT_63969242906879
// MI455X (gfx1250) — compile-verified
//
#include <hip/hip_runtime.h>

#define D 64
#define NEG_SLOPE 0.2f
#define ORD_NEG_INF 0x007FFFFFu  // f2ord(-inf)

typedef float v2f __attribute__((ext_vector_type(2)));
typedef float v8f __attribute__((ext_vector_type(8)));

// ---- helpers -------------------------------------------------------------

// Order-preserving float->uint key: a > b  <=>  f2ord(a) > f2ord(b)
__device__ __forceinline__ unsigned f2ord(float f) {
  unsigned u = __float_as_uint(f);
  return (u & 0x80000000u) ? ~u : (u | 0x80000000u);
}
__device__ __forceinline__ float ord2f(unsigned k) {
  unsigned u = (k & 0x80000000u) ? (k & 0x7fffffffu) : ~k;
  return __uint_as_float(u);
}
__device__ __forceinline__ float lrelu(float v) {
  return v > 0.f ? v : NEG_SLOPE * v;
}
__device__ __forceinline__ float wave_sum(float s) {
#pragma unroll
  for (int off = 16; off; off >>= 1) s += __shfl_xor(s, off, 32);
  return s;
}
__device__ __forceinline__ void edge_sd(const int* __restrict__ ei, int E, int e,
                                        int& s, int& d) {
  if (e < E) { s = ei[e]; d = ei[E + e]; } else { s = e - E; d = e - E; }
}

// ---- init ----------------------------------------------------------------

__global__ __launch_bounds__(256) void gat_init_h(float* __restrict__ h, int n) {
  int i = blockIdx.x * 256 + threadIdx.x;
  if (i < n) h[i] = 0.f;
}

__global__ __launch_bounds__(256) void gat_init_node(
    float* __restrict__ den1, float* __restrict__ den2, float* __restrict__ acc2,
    unsigned* __restrict__ mk1, unsigned* __restrict__ mk2, int n) {
  int i = blockIdx.x * 256 + threadIdx.x;
  if (i < n) {
    den1[i] = 0.f; den2[i] = 0.f; acc2[i] = 0.f;
    mk1[i] = ORD_NEG_INF; mk2[i] = ORD_NEG_INF;
  }
}

// ---- layer-1 dense transforms via V_WMMA_F32_16X16X4_F32 -----------------
// Block = 8 waves. Waves 0..3 compute xl for 4 consecutive 16-row tiles,
// waves 4..7 compute xr for the same tiles. Each wave: 16x64 strip,
// 4 x v8f accumulators, K=64 in steps of 4 (16 iters x 4 WMMA).
// A layout (16x4 f32): lane m=lane&15 holds A[m][kh], A[m][kh+1], kh=(lane>>4)*2.
// B layout (4x16 f32): lane n=lane&15 holds B[kh][n], B[kh+1][n].
// C/D (16x16 f32): VGPR r -> row r (+8 for lanes 16..31), col = lane&15.

__global__ __launch_bounds__(256) void gat_gemm_xlxr(
    const float* __restrict__ x,
    const float* __restrict__ Wl, const float* __restrict__ bl,
    const float* __restrict__ Wr, const float* __restrict__ br,
    float* __restrict__ xl, float* __restrict__ xr, int nRowTiles) {
  int wave = threadIdx.x >> 5;
  int lane = threadIdx.x & 31;
  int rowTile = blockIdx.x * 4 + (wave & 3);
  if (rowTile >= nRowTiles) return;

  const float* W   = (wave & 4) ? Wr : Wl;
  const float* bia = (wave & 4) ? br : bl;
  float*       out = (wave & 4) ? xr : xl;

  const int row0 = rowTile * 16;
  const int m  = lane & 15;          // A row / B col within tile
  const int kh = (lane >> 4) * 2;    // K sub-offset per half-wave

  v8f acc0 = {}, acc1 = {}, acc2 = {}, acc3 = {};
  const float* arow = x + (size_t)(row0 + m) * D + kh;

#pragma unroll
  for (int k = 0; k < D; k += 4) {
    v2f a;
    a.x = arow[k];
    a.y = arow[k + 1];
    const float* wp = W + (size_t)(k + kh) * D + m;  // W[k+kh][m + colTile*16]
    v2f b0; b0.x = wp[0];       b0.y = wp[D];
    v2f b1; b1.x = wp[16];      b1.y = wp[D + 16];
    v2f b2; b2.x = wp[32];      b2.y = wp[D + 32];
    v2f b3; b3.x = wp[48];      b3.y = wp[D + 48];
    acc0 = __builtin_amdgcn_wmma_f32_16x16x4_f32(false, a, false, b0, (short)0, acc0, false, false);
    acc1 = __builtin_amdgcn_wmma_f32_16x16x4_f32(false, a, false, b1, (short)0, acc1, false, false);
    acc2 = __builtin_amdgcn_wmma_f32_16x16x4_f32(false, a, false, b2, (short)0, acc2, false, false);
    acc3 = __builtin_amdgcn_wmma_f32_16x16x4_f32(false, a, false, b3, (short)0, acc3, false, false);
  }

  const int n    = lane & 15;
  const int mOff = (lane >> 4) * 8;
  const float bv0 = bia[n], bv1 = bia[16 + n], bv2 = bia[32 + n], bv3 = bia[48 + n];
#pragma unroll
  for (int r = 0; r < 8; ++r) {
    float* op = out + (size_t)(row0 + mOff + r) * D + n;
    op[0]  = acc0[r] + bv0;
    op[16] = acc1[r] + bv1;
    op[32] = acc2[r] + bv2;
    op[48] = acc3[r] + bv3;
  }
}

// ---- layer-1 edge passes -------------------------------------------------

// score[e] = att . lrelu(xl[src] + xr[dst]); segment max via ordered-uint atomicMax
__global__ __launch_bounds__(256) void gat_edge_score1(
    const int* __restrict__ ei, int E, int Etot,
    const float* __restrict__ xl, const float* __restrict__ xr,
    const float* __restrict__ att,
    float* __restrict__ score, unsigned* __restrict__ mk) {
  int e = blockIdx.x * 8 + (threadIdx.x >> 5);
  if (e >= Etot) return;
  int lane = threadIdx.x & 31;
  int s, d; edge_sd(ei, E, e, s, d);
  const float* ps = xl + (size_t)s * D;
  const float* pd = xr + (size_t)d * D;
  float acc = 0.f;
#pragma unroll
  for (int c = lane; c < D; c += 32)
    acc += att[c] * lrelu(ps[c] + pd[c]);
  acc = wave_sum(acc);
  if (lane == 0) {
    score[e] = acc;
    atomicMax(&mk[d], f2ord(acc));
  }
}

// ex = exp(score - max[dst]); denom[dst] += ex; score <- ex   (shared by both layers)
__global__ __launch_bounds__(256) void gat_edge_exp(
    const int* __restrict__ ei, int E, int Etot,
    float* __restrict__ score, const unsigned* __restrict__ mk,
    float* __restrict__ denom) {
  int e = blockIdx.x * 256 + threadIdx.x;
  if (e >= Etot) return;
  int s, d; edge_sd(ei, E, e, s, d);
  float ex = __expf(score[e] - ord2f(mk[d]));
  score[e] = ex;
  atomicAdd(&denom[d], ex);
}

// h[dst] += (ex/denom[dst]) * xl[src]
__global__ __launch_bounds__(256) void gat_edge_scatter1(
    const int* __restrict__ ei, int E, int Etot,
    const float* __restrict__ xl, const float* __restrict__ ex,
    const float* __restrict__ denom, float* __restrict__ h) {
  int e = blockIdx.x * 8 + (threadIdx.x >> 5);
  if (e >= Etot) return;
  int lane = threadIdx.x & 31;
  int s, d; edge_sd(ei, E, e, s, d);
  float alpha = ex[e] / denom[d];
  const float* ps = xl + (size_t)s * D;
  float*       ph = h  + (size_t)d * D;
#pragma unroll
  for (int c = lane; c < D; c += 32)
    atomicAdd(&ph[c], alpha * ps[c]);
}

// ---- layer-2: collapse to per-node scalars hl/hr (bias1 folded in) -------

__global__ __launch_bounds__(256) void gat_matvec2(
    const float* __restrict__ h, const float* __restrict__ bias1,
    const float* __restrict__ W2l, const float* __restrict__ b2l,
    const float* __restrict__ W2r, const float* __restrict__ b2r,
    float* __restrict__ hl, float* __restrict__ hr, int N) {
  int node = blockIdx.x * 8 + (threadIdx.x >> 5);
  if (node >= N) return;
  int lane = threadIdx.x & 31;
  const float* ph = h + (size_t)node * D;
  float sl = 0.f, sr = 0.f;
#pragma unroll
  for (int c = lane; c < D; c += 32) {
    float v = ph[c] + bias1[c];
    sl += v * W2l[c];
    sr += v * W2r[c];
  }
  sl = wave_sum(sl);
  sr = wave_sum(sr);
  if (lane == 0) {
    hl[node] = sl + b2l[0];
    hr[node] = sr + b2r[0];
  }
}

__global__ __launch_bounds__(256) void gat_edge_score2(
    const int* __restrict__ ei, int E, int Etot,
    const float* __restrict__ hl, const float* __restrict__ hr,
    const float* __restrict__ att2,
    float* __restrict__ score, unsigned* __restrict__ mk) {
  int e = blockIdx.x * 256 + threadIdx.x;
  if (e >= Etot) return;
  int s, d; edge_sd(ei, E, e, s, d);
  float sc = att2[0] * lrelu(hl[s] + hr[d]);
  score[e] = sc;
  atomicMax(&mk[d], f2ord(sc));
}

__global__ __launch_bounds__(256) void gat_edge_scatter2(
    const int* __restrict__ ei, int E, int Etot,
    const float* __restrict__ hl, const float* __restrict__ ex,
    const float* __restrict__ denom, float* __restrict__ acc) {
  int e = blockIdx.x * 256 + threadIdx.x;
  if (e >= Etot) return;
  int s, d; edge_sd(ei, E, e, s, d);
  atomicAdd(&acc[d], (ex[e] / denom[d]) * hl[s]);
}

__global__ __launch_bounds__(256) void gat_finalize(
    const float* __restrict__ acc, const float* __restrict__ bias2,
    float* __restrict__ out, int N) {
  int i = blockIdx.x * 256 + threadIdx.x;
  if (i < N) out[i] = acc[i] + bias2[0];
}

// ---- driver --------------------------------------------------------------

extern "C" void kernel_launch(void* const* d_in, const int* in_sizes, int n_in,
                              void* d_out, int out_size, void* d_ws, size_t ws_size,
                              hipStream_t stream) {
  const float* x     = (const float*)d_in[0];
  const int*   ei    = (const int*)  d_in[1];
  const float* W1l   = (const float*)d_in[2];
  const float* b1l   = (const float*)d_in[3];
  const float* W1r   = (const float*)d_in[4];
  const float* b1r   = (const float*)d_in[5];
  const float* att1  = (const float*)d_in[6];
  const float* bias1 = (const float*)d_in[7];
  const float* W2l   = (const float*)d_in[8];
  const float* b2l   = (const float*)d_in[9];
  const float* W2r   = (const float*)d_in[10];
  const float* b2r   = (const float*)d_in[11];
  const float* att2  = (const float*)d_in[12];
  const float* bias2 = (const float*)d_in[13];
  float* out = (float*)d_out;

  const int N    = in_sizes[0] / D;
  const int E    = in_sizes[1] / 2;
  const int Etot = E + N;     // self loops appended

  // Workspace carve-out (f32 unless noted)
  float* p = (float*)d_ws;
  float* xl  = p; p += (size_t)N * D;
  float* xr  = p; p += (size_t)N * D;
  float* h   = p; p += (size_t)N * D;
  float* sc1 = p; p += Etot;
  float* sc2 = p; p += Etot;
  float* den1 = p; p += N;
  float* den2 = p; p += N;
  float* acc2 = p; p += N;
  float* hl  = p; p += N;
  float* hr  = p; p += N;
  unsigned* mk1 = (unsigned*)p; p += N;
  unsigned* mk2 = (unsigned*)p; p += N;

  const int nRowTiles = (N + 15) / 16;

  gat_init_h<<<(N * D + 255) / 256, 256, 0, stream>>>(h, N * D);
  gat_init_node<<<(N + 255) / 256, 256, 0, stream>>>(den1, den2, acc2, mk1, mk2, N);

  gat_gemm_xlxr<<<(nRowTiles + 3) / 4, 256, 0, stream>>>(
      x, W1l, b1l, W1r, b1r, xl, xr, nRowTiles);

  gat_edge_score1<<<(Etot + 7) / 8, 256, 0, stream>>>(ei, E, Etot, xl, xr, att1, sc1, mk1);
  gat_edge_exp<<<(Etot + 255) / 256, 256, 0, stream>>>(ei, E, Etot, sc1, mk1, den1);
  gat_edge_scatter1<<<(Etot + 7) / 8, 256, 0, stream>>>(ei, E, Etot, xl, sc1, den1, h);

  gat_matvec2<<<(N + 7) / 8, 256, 0, stream>>>(h, bias1, W2l, b2l, W2r, b2r, hl, hr, N);

  gat_edge_score2<<<(Etot + 255) / 256, 256, 0, stream>>>(ei, E, Etot, hl, hr, att2, sc2, mk2);
  gat_edge_exp<<<(Etot + 255) / 256, 256, 0, stream>>>(ei, E, Etot, sc2, mk2, den2);
  gat_edge_scatter2<<<(Etot + 255) / 256, 256, 0, stream>>>(ei, E, Etot, hl, sc2, den2, acc2);

  gat_finalize<<<(N + 255) / 256, 256, 0, stream>>>(acc2, bias2, out, N);
}